// HGCN_37228776522453
// MI455X (gfx1250) — compile-verified
//
#include <hip/hip_runtime.h>
#include <hip/hip_bf16.h>

#define N_NODES 50000
#define FEAT    128
#define EMBED   128
#define HIDDEN  256
#define HOPS    3
#define N_EDGES 800000
#define NNZ     500000
#define N_PAIRS 100000

typedef __attribute__((ext_vector_type(2))) float v2f;
typedef __attribute__((ext_vector_type(8))) float v8f;
typedef __attribute__((ext_vector_type(4))) unsigned int v4u;
typedef __attribute__((ext_vector_type(8))) int v8i;
typedef __attribute__((ext_vector_type(4))) int v4i;

__device__ __forceinline__ void atomicAddF32(float* p, float v) {
    __hip_atomic_fetch_add(p, v, __ATOMIC_RELAXED, __HIP_MEMORY_SCOPE_AGENT);
}

__device__ __forceinline__ float eluf(float x) {
    return x > 0.f ? x : expm1f(x);
}

// ---------------------------------------------------------------------------
// TDM: async 2-D tile load (rows x rowElems f32) from global into LDS with
// per-row padding (row stride in LDS = rowElems + 4 DWORDs -> bank-conflict
// free A-fragment ds_load_b64).  Descriptor per CDNA5 ISA 8.3/8.4.
//   padIntervalCode: 6 -> pad every 128 DW, 7 -> every 256 DW
//   pad amount fixed: code 3 -> 4 DWORDs
// ---------------------------------------------------------------------------
__device__ __forceinline__ void tdm_load_2d_f32(const float* gsrc, unsigned ldsByteOff,
                                                int rowElems, int rows,
                                                int padIntervalCode) {
    unsigned long long ga = (unsigned long long)(const void*)gsrc;
    v4u g0;
    g0.x = 1u;                                            // count=1, user descriptor
    g0.y = ldsByteOff;                                    // lds_addr (bytes)
    g0.z = (unsigned)(ga & 0xFFFFFFFFu);                  // global_addr[31:0]
    g0.w = (unsigned)((ga >> 32) & 0x01FFFFFFu)           // global_addr[56:32]
         | (2u << 30);                                    // type = 2 ("image")
    v8i g1;
    g1[0] = (2 << 16)                                     // data_size = 4B
          | (1 << 20)                                     // pad_enable
          | (padIntervalCode << 22)                       // pad_interval
          | (3 << 25);                                    // pad_amount = 4 DWORDs
    g1[1] = (rowElems & 0xFFFF) << 16;                    // tensor_dim0[15:0]
    g1[2] = ((rowElems >> 16) & 0xFFFF)                   // tensor_dim0[31:16]
          | ((rows & 0xFFFF) << 16);                      // tensor_dim1[15:0]
    g1[3] = ((rows >> 16) & 0xFFFF)                       // tensor_dim1[31:16]
          | ((rowElems & 0xFFFF) << 16);                  // tile_dim0
    g1[4] = rows & 0xFFFF;                                // tile_dim1 (tile_dim2=0 -> 2D)
    g1[5] = rowElems;                                     // tensor_dim0_stride[31:0]
    g1[6] = 0;                                            // stride hi / dim1_stride lo
    g1[7] = 0;
    v4i z4 = {0, 0, 0, 0};
    v8i z8 = {0, 0, 0, 0, 0, 0, 0, 0};
    __builtin_amdgcn_tensor_load_to_lds(g0, g1, z4, z4, z8, 0);
}

// ---------------------------------------------------------------------------
// 1) Scatter sparse feature matrix into dense x [N_NODES, FEAT]
// ---------------------------------------------------------------------------
__global__ void scatter_feat_kernel(const int* __restrict__ fr,
                                    const int* __restrict__ fc,
                                    const float* __restrict__ fv,
                                    float* __restrict__ x, int nnz) {
    int i = blockIdx.x * blockDim.x + threadIdx.x;
    if (i < nnz) atomicAddF32(&x[fr[i] * FEAT + fc[i]], fv[i]);
}

// ---------------------------------------------------------------------------
// 2) SpMM: out += A @ h  (one wave32 per edge; lane handles 4 contiguous feats)
//    out pre-zeroed. 102M no-return f32 atomics land in L2 (25.6MB << 192MB).
// ---------------------------------------------------------------------------
__global__ void spmm_kernel(const int* __restrict__ rows,
                            const int* __restrict__ cols,
                            const float* __restrict__ vals,
                            const float* __restrict__ h,
                            float* __restrict__ out, int nEdges) {
    int e = blockIdx.x * (blockDim.x >> 5) + (threadIdx.x >> 5);
    if (e >= nEdges) return;
    int lane = threadIdx.x & 31;
    int r = rows[e], c = cols[e];
    float v = vals[e];
    float4 x = *(const float4*)(h + c * FEAT + lane * 4);
    float* dst = out + r * FEAT + lane * 4;
    atomicAddF32(dst + 0, x.x * v);
    atomicAddF32(dst + 1, x.y * v);
    atomicAddF32(dst + 2, x.z * v);
    atomicAddF32(dst + 3, x.w * v);
}

// ---------------------------------------------------------------------------
// 3) Per-hop dense GEMM + ReLU with fp32 WMMA:
//    out[N,128] = relu(h[N,128] @ W[128,128] + bias[128])
//    Block = 8 waves over 16 rows x 128 cols. A strip (16x128) staged in LDS
//    by the TDM once per block (row stride 132 DW, conflict-free).
// ---------------------------------------------------------------------------
#define HOP_LDS_STRIDE 132   // 128 + 4 pad DWORDs
__global__ void hop_gemm_relu_kernel(const float* __restrict__ h,
                                     const float* __restrict__ W,
                                     const float* __restrict__ bias,
                                     float* __restrict__ out) {
    __shared__ float ldsA[16 * HOP_LDS_STRIDE];
    const int wave = threadIdx.x >> 5;
    const int lane = threadIdx.x & 31;
    const int m    = lane & 15;
    const int hi   = lane >> 4;          // 0 or 1
    const int rowBase = blockIdx.x * 16;
    const int colBase = wave * 16;

    if (threadIdx.x < 32) {              // one wave drives the TDM
        tdm_load_2d_f32(h + (size_t)rowBase * FEAT,
                        (unsigned)(unsigned long long)(uintptr_t)ldsA,
                        FEAT, 16, /*pad every 128 DW*/ 6);
        __builtin_amdgcn_s_wait_tensorcnt(0);
    }
    __syncthreads();

    v8f acc = {};
    const float* arow = &ldsA[m * HOP_LDS_STRIDE];
    #pragma unroll
    for (int k = 0; k < FEAT; k += 4) {
        const int ka = k + 2 * hi;
        v2f a = *(const v2f*)(arow + ka);            // ds_load_b64, A: M=m, K=ka,ka+1
        v2f b;
        b.x = W[ka * EMBED + colBase + m];           // B: K=ka,   N=colBase+m
        b.y = W[(ka + 1) * EMBED + colBase + m];     // B: K=ka+1, N=colBase+m
        acc = __builtin_amdgcn_wmma_f32_16x16x4_f32(
            false, a, false, b, (short)0, acc, false, false);
    }
    const float bv = bias[colBase + m];
    #pragma unroll
    for (int r = 0; r < 8; ++r) {
        float v = acc[r] + bv;
        v = v > 0.f ? v : 0.f;
        out[(rowBase + r + 8 * hi) * EMBED + colBase + m] = v;
    }
}

// ---------------------------------------------------------------------------
// 4) Hop attention: scores -> softmax over 3 hops -> enhanced [N,128]
// ---------------------------------------------------------------------------
__global__ void attn_kernel(const float* __restrict__ f0,
                            const float* __restrict__ f1,
                            const float* __restrict__ f2,
                            const float* __restrict__ aw,   // [HOPS,128]
                            float* __restrict__ enh) {
    const int wave = threadIdx.x >> 5;
    const int lane = threadIdx.x & 31;
    const int n = blockIdx.x * 8 + wave;           // 50000/8 = 6250 exact
    const int f = lane * 4;
    float4 x0 = *(const float4*)(f0 + n * EMBED + f);
    float4 x1 = *(const float4*)(f1 + n * EMBED + f);
    float4 x2 = *(const float4*)(f2 + n * EMBED + f);
    float4 w0 = *(const float4*)(aw + 0 * EMBED + f);
    float4 w1 = *(const float4*)(aw + 1 * EMBED + f);
    float4 w2 = *(const float4*)(aw + 2 * EMBED + f);
    float s0 = x0.x*w0.x + x0.y*w0.y + x0.z*w0.z + x0.w*w0.w;
    float s1 = x1.x*w1.x + x1.y*w1.y + x1.z*w1.z + x1.w*w1.w;
    float s2 = x2.x*w2.x + x2.y*w2.y + x2.z*w2.z + x2.w*w2.w;
    #pragma unroll
    for (int off = 16; off > 0; off >>= 1) {
        s0 += __shfl_xor(s0, off, 32);
        s1 += __shfl_xor(s1, off, 32);
        s2 += __shfl_xor(s2, off, 32);
    }
    float mx = fmaxf(s0, fmaxf(s1, s2));
    float e0 = expf(s0 - mx), e1 = expf(s1 - mx), e2 = expf(s2 - mx);
    float inv = 1.f / (e0 + e1 + e2);
    float a0 = e0 * inv, a1 = e1 * inv, a2 = e2 * inv;
    float4 o;
    o.x = a0*x0.x + a1*x1.x + a2*x2.x;
    o.y = a0*x0.y + a1*x1.y + a2*x2.y;
    o.z = a0*x0.z + a1*x1.z + a2*x2.z;
    o.w = a0*x0.w + a1*x1.w + a2*x2.w;
    *(float4*)(enh + n * EMBED + f) = o;
}

// ---------------------------------------------------------------------------
// 5) Pair fusion: fused[p] = elu([|f1-f2| , f1*f2]) -> d_out fused region
// ---------------------------------------------------------------------------
__global__ void fuse_kernel(const float* __restrict__ enh,
                            const int* __restrict__ idx0,
                            const int* __restrict__ idx1,
                            float* __restrict__ fused) {
    const int wave = threadIdx.x >> 5;
    const int lane = threadIdx.x & 31;
    const int p = blockIdx.x * 8 + wave;           // 100000/8 = 12500 exact
    const int f = lane * 4;
    const int i0 = idx0[p], i1 = idx1[p];
    float4 a = *(const float4*)(enh + i0 * EMBED + f);
    float4 b = *(const float4*)(enh + i1 * EMBED + f);
    float4 d, q;
    d.x = eluf(fabsf(a.x - b.x)); q.x = eluf(a.x * b.x);
    d.y = eluf(fabsf(a.y - b.y)); q.y = eluf(a.y * b.y);
    d.z = eluf(fabsf(a.z - b.z)); q.z = eluf(a.z * b.z);
    d.w = eluf(fabsf(a.w - b.w)); q.w = eluf(a.w * b.w);
    *(float4*)(fused + p * (2 * EMBED) + f) = d;
    *(float4*)(fused + p * (2 * EMBED) + EMBED + f) = q;
}

// ---------------------------------------------------------------------------
// 6) Fused decoder: hid = elu(fused @ W1 + b1); logits = hid @ W2 + b2
//    Block covers 16 pairs x 256 cols; 8 waves x 2 col tiles; K=256 ->
//    64 steps x 2 WMMA. A strip (16x256) staged in LDS via TDM (stride 260).
//    Row-dot with W2 reduced via LDS f32 atomics.
// ---------------------------------------------------------------------------
#define DEC_LDS_STRIDE 260   // 256 + 4 pad DWORDs
__global__ void decoder_kernel(const float* __restrict__ fused,   // [P,256]
                               const float* __restrict__ W1,      // [256,256]
                               const float* __restrict__ b1,      // [256]
                               const float* __restrict__ W2,      // [256]
                               const float* __restrict__ b2,      // [1]
                               float* __restrict__ logits) {      // [P]
    __shared__ float ldsA[16 * DEC_LDS_STRIDE];
    __shared__ float rowsum[16];
    const int wave = threadIdx.x >> 5;
    const int lane = threadIdx.x & 31;
    const int m  = lane & 15;
    const int hi = lane >> 4;
    const int pairBase = blockIdx.x * 16;          // 100000/16 = 6250 exact
    const int c0 = wave * 32, c1 = wave * 32 + 16;

    if (threadIdx.x < 16) rowsum[threadIdx.x] = 0.f;
    if (threadIdx.x < 32) {              // one wave drives the TDM
        tdm_load_2d_f32(fused + (size_t)pairBase * (2 * EMBED),
                        (unsigned)(unsigned long long)(uintptr_t)ldsA,
                        2 * EMBED, 16, /*pad every 256 DW*/ 7);
        __builtin_amdgcn_s_wait_tensorcnt(0);
    }
    __syncthreads();

    v8f acc0 = {}, acc1 = {};
    const float* arow = &ldsA[m * DEC_LDS_STRIDE];
    #pragma unroll 8
    for (int k = 0; k < 2 * EMBED; k += 4) {
        const int ka = k + 2 * hi;
        v2f a = *(const v2f*)(arow + ka);          // ds_load_b64
        v2f bA, bB;
        bA.x = W1[ka * HIDDEN + c0 + m];
        bA.y = W1[(ka + 1) * HIDDEN + c0 + m];
        bB.x = W1[ka * HIDDEN + c1 + m];
        bB.y = W1[(ka + 1) * HIDDEN + c1 + m];
        acc0 = __builtin_amdgcn_wmma_f32_16x16x4_f32(
            false, a, false, bA, (short)0, acc0, false, false);
        acc1 = __builtin_amdgcn_wmma_f32_16x16x4_f32(
            false, a, false, bB, (short)0, acc1, false, false);
    }
    const float bias0 = b1[c0 + m], bias1 = b1[c1 + m];
    const float w2a = W2[c0 + m],  w2b = W2[c1 + m];
    #pragma unroll
    for (int r = 0; r < 8; ++r) {
        float h0 = eluf(acc0[r] + bias0);
        float h1 = eluf(acc1[r] + bias1);
        atomicAdd(&rowsum[r + 8 * hi], h0 * w2a + h1 * w2b);
    }
    __syncthreads();
    if (threadIdx.x < 16)
        logits[pairBase + threadIdx.x] = rowsum[threadIdx.x] + b2[0];
}

// ---------------------------------------------------------------------------
// Host launch. Workspace layout (floats), total 6 * 25.6MB = 153.6MB:
//   hA, hB (h ping/pong), feats[3], enhanced
// d_out: logits [100000] then fused [100000*256].
// ---------------------------------------------------------------------------
extern "C" void kernel_launch(void* const* d_in, const int* in_sizes, int n_in,
                              void* d_out, int out_size, void* d_ws, size_t ws_size,
                              hipStream_t stream) {
    const int*   adj_rows = (const int*)  d_in[0];
    const int*   adj_cols = (const int*)  d_in[1];
    const float* adj_vals = (const float*)d_in[2];
    const int*   feat_rows= (const int*)  d_in[3];
    const int*   feat_cols= (const int*)  d_in[4];
    const float* feat_vals= (const float*)d_in[5];
    const int*   idx      = (const int*)  d_in[6];   // [2, N_PAIRS]
    const float* W        = (const float*)d_in[7];   // [HOPS,128,128]
    const float* b        = (const float*)d_in[8];   // [HOPS,128]
    const float* attn_w   = (const float*)d_in[9];   // [HOPS,128]
    const float* dec_W1   = (const float*)d_in[10];  // [256,256]
    const float* dec_b1   = (const float*)d_in[11];  // [256]
    const float* dec_W2   = (const float*)d_in[12];  // [256,1]
    const float* dec_b2   = (const float*)d_in[13];  // [1]

    float* ws = (float*)d_ws;
    const size_t NF = (size_t)N_NODES * FEAT;
    float* hA  = ws;
    float* hB  = ws + NF;
    float* feats[HOPS] = { ws + 2 * NF, ws + 3 * NF, ws + 4 * NF };
    float* enh = ws + 5 * NF;

    float* logits = (float*)d_out;
    float* fused  = (float*)d_out + N_PAIRS;

    // densify features into hA
    (void)hipMemsetAsync(hA, 0, NF * sizeof(float), stream);
    scatter_feat_kernel<<<(NNZ + 255) / 256, 256, 0, stream>>>(
        feat_rows, feat_cols, feat_vals, hA, NNZ);

    // 3 hops: spmm + gemm/relu
    float* cur = hA;
    float* nxt = hB;
    for (int hop = 0; hop < HOPS; ++hop) {
        (void)hipMemsetAsync(nxt, 0, NF * sizeof(float), stream);
        spmm_kernel<<<N_EDGES / 8, 256, 0, stream>>>(
            adj_rows, adj_cols, adj_vals, cur, nxt, N_EDGES);
        hop_gemm_relu_kernel<<<N_NODES / 16, 256, 0, stream>>>(
            nxt, W + (size_t)hop * FEAT * EMBED, b + hop * EMBED, feats[hop]);
        float* t = cur; cur = nxt; nxt = t;
    }

    // attention blend
    attn_kernel<<<N_NODES / 8, 256, 0, stream>>>(
        feats[0], feats[1], feats[2], attn_w, enh);

    // pair fusion (writes output "fused" region, also decoder input)
    fuse_kernel<<<N_PAIRS / 8, 256, 0, stream>>>(
        enh, idx, idx + N_PAIRS, fused);

    // fused decoder GEMM + elu + dot(W2)
    decoder_kernel<<<N_PAIRS / 16, 256, 0, stream>>>(
        fused, dec_W1, dec_b1, dec_W2, dec_b2, logits);
}